// ReadoutPhase_37606733644085
// MI455X (gfx1250) — compile-verified
//
#include <hip/hip_runtime.h>
#include <math.h>

// MI455X / gfx1250, wave32. One 32-thread block per output graph segment.
// Memory-bound: reads x exactly once (~1.03 GB @ 23.3 TB/s ~= 44 us floor).
// Scores via chained V_WMMA_F32_16X16X4_F32; x streamed global->LDS with the
// CDNA5 async-to-LDS DMA path (double-buffered) when the builtin is available.

typedef __attribute__((ext_vector_type(2))) float v2f;
typedef __attribute__((ext_vector_type(4))) float v4f;
typedef __attribute__((ext_vector_type(8))) float v8f;
typedef __attribute__((ext_vector_type(4))) int   v4i;

#define GDIM 256
#define TSTRIDE 260  // row*260 % 64 == row*4 -> conflict-free A-repack; 260*4 B is 16B-aligned

#if defined(__has_builtin)
#  if __has_builtin(__builtin_amdgcn_global_load_async_to_lds_b128)
#    define USE_ASYNC_LDS 1
#  endif
#endif
#ifndef USE_ASYNC_LDS
#  define USE_ASYNC_LDS 0
#endif

// AS(1) = global ("__device__" in clang HIP diagnostics), AS(3) = LDS
typedef __attribute__((address_space(1))) v4i as1_v4i;
typedef __attribute__((address_space(3))) v4i as3_v4i;

__device__ __forceinline__ void async_wait_all() {
#if defined(__has_builtin)
#  if __has_builtin(__builtin_amdgcn_s_wait_asynccnt)
    __builtin_amdgcn_s_wait_asynccnt(0);
#  else
    asm volatile("s_wait_asynccnt 0x0" ::: "memory");
#  endif
#else
    asm volatile("s_wait_asynccnt 0x0" ::: "memory");
#endif
    asm volatile("" ::: "memory");  // compiler ordering fence
}

__global__ __launch_bounds__(32)
void gated_readout_kernel(const float* __restrict__ x,
                          const int*   __restrict__ batch,   // sorted segment ids
                          const float* __restrict__ W,
                          const float* __restrict__ bptr,
                          float* __restrict__ out,
                          int N)
{
    __shared__ __align__(16) float sW[GDIM];
    __shared__ __align__(16) float tile[2][16][TSTRIDE];   // double-buffered 16x256 tile

    const int g    = blockIdx.x;
    const int lane = threadIdx.x;
    const int hlf  = lane >> 4;    // 0: lanes 0-15 (K0,K1), 1: lanes 16-31 (K2,K3)
    const int mrow = lane & 15;    // M row this lane feeds into the A matrix

    for (int i = lane; i < GDIM; i += 32) sW[i] = W[i];
    const float bias = bptr[0];

    // ---- segment range [start, end) via binary search over sorted batch ----
    int lo = 0, hi = N;
    while (lo < hi) { int mid = (lo + hi) >> 1; if (batch[mid] < g) lo = mid + 1; else hi = mid; }
    const int start = lo;
    hi = N;
    while (lo < hi) { int mid = (lo + hi) >> 1; if (batch[mid] <= g) lo = mid + 1; else hi = mid; }
    const int end = lo;

    float accs[8], accm[8];
#pragma unroll
    for (int j = 0; j < 8; ++j) { accs[j] = 0.0f; accm[j] = -INFINITY; }

    const int nt = (end - start + 15) >> 4;   // 16-node tiles in this segment

#if USE_ASYNC_LDS
    // Issue one 16KB tile as 32 per-lane async b128 DMA ops (ASYNCcnt <= 32).
    // Addresses are clamped in-bounds; tail rows get zeroed after the wait.
    auto issue_tile = [&](int buf, int n0) {
#pragma unroll 4
        for (int it = 0; it < 32; ++it) {
            const int i   = lane + (it << 5);
            const int row = i >> 6;
            const int c4  = (i & 63) << 2;
            const int node = min(n0 + row, N - 1);
            const float* src = x + (size_t)node * GDIM + c4;
            __builtin_amdgcn_global_load_async_to_lds_b128(
                (as1_v4i*)(void*)src,
                (as3_v4i*)(void*)&tile[buf][row][c4], 0, 0);
        }
    };
    if (nt > 0) issue_tile(0, start);
#endif

    __syncthreads();  // sW visible (single wave: cheap)

    for (int t = 0; t < nt; ++t) {
        const int n0  = start + (t << 4);
        const int nv  = (end - n0 < 16) ? (end - n0) : 16;
        const int buf = t & 1;

#if USE_ASYNC_LDS
        async_wait_all();                    // tile t landed in LDS
        if (nv < 16) {                       // zero-pad tail rows (last tile only)
            for (int i = lane; i < ((16 - nv) << 6); i += 32) {
                const int row = nv + (i >> 6);
                const int c4  = (i & 63) << 2;
                *(v4f*)&tile[buf][row][c4] = (v4f)(0.0f);
            }
        }
        if (t + 1 < nt) issue_tile(buf ^ 1, n0 + 16);   // overlap DMA with compute
        asm volatile("" ::: "memory");
#else
        // synchronous fallback: coalesced non-temporal b128 loads via VGPRs
#pragma unroll 4
        for (int it = 0; it < 32; ++it) {
            const int i   = lane + (it << 5);
            const int row = i >> 6;
            const int c4  = (i & 63) << 2;
            v4f v = (v4f)(0.0f);
            if (row < nv)
                v = __builtin_nontemporal_load((const v4f*)(x + (size_t)(n0 + row) * GDIM + c4));
            *(v4f*)&tile[buf][row][c4] = v;
        }
        __syncthreads();
#endif

        // ---- scores for 16 nodes via chained V_WMMA_F32_16X16X4_F32 ----
        // A[m][k] = x[node m][4*ch + k] ; B[k][n] = W[4*ch + k] for all n
        // => D[m][n] = dot(x_m, W) replicated across n after 64 chunks.
        v8f c = {};
#pragma unroll 8
        for (int ch = 0; ch < 64; ++ch) {
            const int kb = (ch << 2) + (hlf << 1);
            const float2 av = *(const float2*)&tile[buf][mrow][kb]; // conflict-free b64
            const float2 bv = *(const float2*)&sW[kb];              // LDS broadcast
            v2f a; a[0] = av.x; a[1] = av.y;
            v2f b; b[0] = bv.x; b[1] = bv.y;
            c = __builtin_amdgcn_wmma_f32_16x16x4_f32(false, a, false, b,
                                                      (short)0, c, false, false);
        }

        // ---- bias + fast sigmoid (v_exp_f32 + v_rcp_f32); shfl_xor(16) swap ----
        // lanes 0-15: c[r] = score(M=r) ; lanes 16-31: c[r] = score(M=8+r)
        float sloc[8], soth[8];
#pragma unroll
        for (int r = 0; r < 8; ++r) {
            const float z = c[r] + bias;
            sloc[r] = __builtin_amdgcn_rcpf(1.0f + __expf(-z));
            soth[r] = __shfl_xor(sloc[r], 16, 32);
        }

        // ---- gated sum + max; lane owns dims [8*lane, 8*lane+8) ----
        const int d0 = lane << 3;
#pragma unroll
        for (int m = 0; m < 16; ++m) {
            if (m < nv) {
                const float sm = (m < 8) ? (hlf ? soth[m] : sloc[m])
                                         : (hlf ? sloc[m - 8] : soth[m - 8]);
                const v4f xa = *(const v4f*)&tile[buf][m][d0];
                const v4f xb = *(const v4f*)&tile[buf][m][d0 + 4];
                accs[0] = fmaf(sm, xa.x, accs[0]); accm[0] = fmaxf(accm[0], xa.x);
                accs[1] = fmaf(sm, xa.y, accs[1]); accm[1] = fmaxf(accm[1], xa.y);
                accs[2] = fmaf(sm, xa.z, accs[2]); accm[2] = fmaxf(accm[2], xa.z);
                accs[3] = fmaf(sm, xa.w, accs[3]); accm[3] = fmaxf(accm[3], xa.w);
                accs[4] = fmaf(sm, xb.x, accs[4]); accm[4] = fmaxf(accm[4], xb.x);
                accs[5] = fmaf(sm, xb.y, accs[5]); accm[5] = fmaxf(accm[5], xb.y);
                accs[6] = fmaf(sm, xb.z, accs[6]); accm[6] = fmaxf(accm[6], xb.z);
                accs[7] = fmaf(sm, xb.w, accs[7]); accm[7] = fmaxf(accm[7], xb.w);
            }
        }

#if !USE_ASYNC_LDS
        __syncthreads();   // protect tile WAR for next iteration
#endif
    }

    // ---- write out: [G, 512] = concat(sum[256], max[256]) ----
    float* o = out + (size_t)g * (2 * GDIM);
    const int d0 = lane << 3;
    v4f s0; s0.x = accs[0]; s0.y = accs[1]; s0.z = accs[2]; s0.w = accs[3];
    v4f s1; s1.x = accs[4]; s1.y = accs[5]; s1.z = accs[6]; s1.w = accs[7];
    v4f m0; m0.x = accm[0]; m0.y = accm[1]; m0.z = accm[2]; m0.w = accm[3];
    v4f m1; m1.x = accm[4]; m1.y = accm[5]; m1.z = accm[6]; m1.w = accm[7];
    *(v4f*)&o[d0]            = s0;
    *(v4f*)&o[d0 + 4]        = s1;
    *(v4f*)&o[GDIM + d0]     = m0;
    *(v4f*)&o[GDIM + d0 + 4] = m1;
}

extern "C" void kernel_launch(void* const* d_in, const int* in_sizes, int n_in,
                              void* d_out, int out_size, void* d_ws, size_t ws_size,
                              hipStream_t stream) {
    const float* x     = (const float*)d_in[0];
    const int*   batch = (const int*)d_in[1];   // harness materializes integer inputs as int32
    const float* W     = (const float*)d_in[2];
    const float* b     = (const float*)d_in[3];
    float*       out   = (float*)d_out;

    const int N = in_sizes[1];              // number of nodes (len(batch))
    const int G = out_size / (2 * GDIM);

    gated_readout_kernel<<<G, 32, 0, stream>>>(x, batch, W, b, out, N);
}